// SpatialTransformer_34342558499140
// MI455X (gfx1250) — compile-verified
//
#include <hip/hip_runtime.h>

// 3D SpatialTransformer (grid_sample, trilinear, zeros padding, align_corners=True)
// src[B=2,C=2,160,160,160], flow[B,3,160,160,160], out[B,C,160,160,160], fp32.
//
// MI455X strategy: bandwidth-bound gather, no matrix structure -> no WMMA.
//  * flow/out streamed once -> non-temporal (TH=NT); src gathers regular
//    temporal so the 65.5 MB src volume stays resident in the 192 MB L2.
//  * one thread per spatial voxel computes BOTH channels (shared flow loads,
//    index math, weights); channel-1 gather is idx + N.
//  * launch geometry (W threads, grid=(H,D,B)) makes (b,d,h,w) free:
//    no per-thread divides, no bounds check; h/d/b are wave-uniform so the
//    base-address math lands in SALU/SGPRs.
//  * branchless zeros-padding: clamp coords (loads stay in-range), zero the
//    weight when out of range.

namespace {

constexpr int B = 2;
constexpr int C = 2;
constexpr int D = 160;
constexpr int H = 160;
constexpr int W = 160;
constexpr int N = D * H * W;  // 4,096,000 voxels per volume

__global__ __launch_bounds__(W) void st3d_kernel(
    const float* __restrict__ src, const float* __restrict__ flow,
    float* __restrict__ out) {
  const int w = threadIdx.x;   // 0..159 (5 wave32 waves)
  const int h = blockIdx.x;    // 0..159  (wave-uniform)
  const int d = blockIdx.y;    // 0..159  (wave-uniform)
  const int b = blockIdx.z;    // 0..1    (wave-uniform)

  const int s = (d * H + h) * W + w;  // spatial index within one volume

  // Flow components at stride N: [B,3,D,H,W]. Streamed once -> NT loads.
  const float* f = flow + (size_t)b * (3 * N) + s;
  const float z = (float)d + __builtin_nontemporal_load(f);
  const float y = (float)h + __builtin_nontemporal_load(f + N);
  const float x = (float)w + __builtin_nontemporal_load(f + 2 * N);

  const float z0f = floorf(z);
  const float y0f = floorf(y);
  const float x0f = floorf(x);
  const float fz = z - z0f;
  const float fy = y - y0f;
  const float fx = x - x0f;
  const int z0 = (int)z0f;
  const int y0 = (int)y0f;
  const int x0 = (int)x0f;

  const float wz0 = 1.0f - fz;
  const float wy0 = 1.0f - fy;
  const float wx0 = 1.0f - fx;

  const float* sb = src + (size_t)b * (C * N);  // channel-0 base (uniform)

  float acc0 = 0.0f;  // channel 0
  float acc1 = 0.0f;  // channel 1

#pragma unroll
  for (int dz = 0; dz < 2; ++dz) {
    const int zi = z0 + dz;
    const bool vz = (unsigned)zi < (unsigned)D;
    const int zc = min(max(zi, 0), D - 1);
    const float wz = dz ? fz : wz0;
#pragma unroll
    for (int dy = 0; dy < 2; ++dy) {
      const int yi = y0 + dy;
      const bool vy = (unsigned)yi < (unsigned)H;
      const int yc = min(max(yi, 0), H - 1);
      const float wzy = wz * (dy ? fy : wy0);
      const int base = (zc * H + yc) * W;
#pragma unroll
      for (int dx = 0; dx < 2; ++dx) {
        const int xi = x0 + dx;
        const bool v = vz & vy & ((unsigned)xi < (unsigned)W);
        const int xc = min(max(xi, 0), W - 1);
        // zeros padding: clamped in-range load, weight forced to 0 if OOB.
        const float ww = v ? wzy * (dx ? fx : wx0) : 0.0f;
        const int idx = base + xc;
        const float v0 = sb[idx];      // channel 0 gather (temporal: L2-resident)
        const float v1 = sb[idx + N];  // channel 1 gather (same address + N)
        acc0 = fmaf(ww, v0, acc0);
        acc1 = fmaf(ww, v1, acc1);
      }
    }
  }

  // out[b, c, s]; streamed once -> NT stores.
  float* o = out + (size_t)b * (C * N) + s;
  __builtin_nontemporal_store(acc0, o);
  __builtin_nontemporal_store(acc1, o + N);
}

}  // namespace

extern "C" void kernel_launch(void* const* d_in, const int* in_sizes, int n_in,
                              void* d_out, int out_size, void* d_ws,
                              size_t ws_size, hipStream_t stream) {
  (void)in_sizes; (void)n_in; (void)d_ws; (void)ws_size; (void)out_size;
  const float* src = (const float*)d_in[0];
  const float* flow = (const float*)d_in[1];
  float* out = (float*)d_out;

  dim3 block(W, 1, 1);   // 160 threads = 5 wave32 waves
  dim3 grid(H, D, B);    // (160, 160, 2) blocks; one W-row each
  st3d_kernel<<<grid, block, 0, stream>>>(src, flow, out);
}